// Actor_50276887167256
// MI455X (gfx1250) — compile-verified
//
#include <hip/hip_runtime.h>
#include <hip/hip_bf16.h>

typedef float v2f __attribute__((ext_vector_type(2)));
typedef float v8f __attribute__((ext_vector_type(8)));

#define NN    50000
#define EE    800000
#define DIN   256
#define HD    128
#define AD    32

// ---------- helpers: monotonic float<->uint ordering for atomic max ----------
__device__ __forceinline__ unsigned f2ord(float f) {
    unsigned u = __float_as_uint(f);
    return (u & 0x80000000u) ? ~u : (u | 0x80000000u);
}
__device__ __forceinline__ float ord2f(unsigned u) {
    unsigned b = (u & 0x80000000u) ? (u & 0x7FFFFFFFu) : ~u;
    return __uint_as_float(b);
}

// ---------------------------------------------------------------------------
// Fused WMMA GEMM:  Out[nrows x NCOL] = act( transformA(A[nrows x K]) @ B[K x NCOL] + bias )
// AMODE 0: A as-is
// AMODE 1: a = relu(a + t0[k]) * t1[k] + t2[k]   (post-GAT relu + BN0 fused on load)
// AMODE 2: a = a * t0[k] + t1[k]                 (BN2 fused on load)
// Each wave: 16 rows x NCOL cols via NT = NCOL/16 accumulator tiles of v_wmma_f32_16x16x4_f32.
// B staged through LDS in KC x NCOL chunks, PAIR-MAJOR swizzled so each lane's
// {B[k][n], B[k+1][n]} fragment is one aligned ds_load_b64 straight into the
// even-aligned VGPR pair WMMA SRC1 wants (no re-pairing movs).
//   Bs[((k>>1)*NCOL + n)*2 + (k&1)] = B[kc+k][n]
// ---------------------------------------------------------------------------
template<int K, int NCOL, int AMODE, bool RELU_OUT>
__global__ __launch_bounds__(256) void gemm_wmma_kernel(
    const float* __restrict__ A, const float* __restrict__ B,
    const float* __restrict__ bias,
    const float* __restrict__ t0, const float* __restrict__ t1, const float* __restrict__ t2,
    float* __restrict__ Out, int nrows)
{
    constexpr int KC = 64;
    constexpr int NT = NCOL / 16;
    __shared__ float Bs[KC * NCOL];

    const int lane    = threadIdx.x & 31;
    const int wave    = threadIdx.x >> 5;
    const int rowBase = (blockIdx.x * 8 + wave) * 16;
    const int r       = lane & 15;          // row-within-tile (A) / col-within-tile (B, C/D)
    const int kk      = (lane >> 4) << 1;   // k-pair selector for A/B fragments (0 or 2)
    const bool active = rowBase < nrows;    // uniform per wave (nrows multiple of 16)

    v8f acc[NT] = {};

    for (int kc = 0; kc < K; kc += KC) {
        // cooperative stage of B chunk into LDS (pair-major swizzle)
        for (int idx = threadIdx.x; idx < KC * NCOL; idx += 256) {
            const int k = idx / NCOL;
            const int n = idx % NCOL;
            Bs[(((k >> 1) * NCOL) + n) * 2 + (k & 1)] = B[(size_t)(kc + k) * NCOL + n];
        }
        __syncthreads();

        if (active) {
            const float* arow = A + (size_t)(rowBase + r) * K;
#pragma unroll
            for (int k0 = 0; k0 < KC; k0 += 4) {
                const int kidx = kc + k0 + kk;
                float x0 = arow[kidx];
                float x1 = arow[kidx + 1];
                if (AMODE == 1) {
                    x0 = fmaxf(x0 + t0[kidx],     0.f) * t1[kidx]     + t2[kidx];
                    x1 = fmaxf(x1 + t0[kidx + 1], 0.f) * t1[kidx + 1] + t2[kidx + 1];
                } else if (AMODE == 2) {
                    x0 = x0 * t0[kidx]     + t1[kidx];
                    x1 = x1 * t0[kidx + 1] + t1[kidx + 1];
                }
                v2f a; a.x = x0; a.y = x1;
                const int kpBase = ((k0 + kk) >> 1) * NCOL;   // pair row base in LDS
#pragma unroll
                for (int j = 0; j < NT; ++j) {
                    const v2f b = *(const v2f*)&Bs[(kpBase + j * 16 + r) * 2];
                    acc[j] = __builtin_amdgcn_wmma_f32_16x16x4_f32(
                        false, a, false, b, (short)0, acc[j], false, false);
                }
            }
        }
        __syncthreads();
    }

    if (active) {
        const int rsub = (lane >> 4) * 8;  // C/D layout: vgpr i -> M=i (lanes 0-15) / M=i+8 (16-31)
#pragma unroll
        for (int j = 0; j < NT; ++j) {
            const int col = j * 16 + r;
            const float bv = bias ? bias[col] : 0.f;
#pragma unroll
            for (int i = 0; i < 8; ++i) {
                float v = acc[j][i] + bv;
                if (RELU_OUT) v = fmaxf(v, 0.f);
                Out[(size_t)(rowBase + rsub + i) * NCOL + col] = v;
            }
        }
    }
}

// ---------------------------------------------------------------------------
// e_src[i] = h[i,:] . a_src ; e_dst[i] = h[i,:] . a_dst
// ---------------------------------------------------------------------------
__global__ __launch_bounds__(256) void node_attn_kernel(
    const float* __restrict__ h, const float* __restrict__ a_src,
    const float* __restrict__ a_dst,
    float* __restrict__ e_src, float* __restrict__ e_dst, int n)
{
    const int row = blockIdx.x * 256 + threadIdx.x;
    if (row >= n) return;
    const float* hr = h + (size_t)row * HD;
    float ss = 0.f, sd = 0.f;
#pragma unroll 8
    for (int k = 0; k < HD; ++k) {
        const float v = hr[k];
        ss += v * a_src[k];
        sd += v * a_dst[k];
    }
    e_src[row] = ss;
    e_dst[row] = sd;
}

// edge pass 1: e = leaky_relu(e_src[s] + e_dst[d]); segment max into m_ord[d]
__global__ __launch_bounds__(256) void edge_pass1(
    const long long* __restrict__ ei, const float* __restrict__ es,
    const float* __restrict__ ed, float* __restrict__ e_edge,
    unsigned* __restrict__ m_ord, int nE)
{
    const int i = blockIdx.x * 256 + threadIdx.x;
    if (i >= nE) return;
    const int s = (int)ei[i];
    const int d = (int)ei[(size_t)nE + i];
    float e = es[s] + ed[d];
    e = (e > 0.f) ? e : 0.2f * e;
    e_edge[i] = e;
    atomicMax(&m_ord[d], f2ord(e));
}

// edge pass 2: w = exp(e - m[d]) (stored in place); denom[d] += w
__global__ __launch_bounds__(256) void edge_pass2(
    const long long* __restrict__ ei, float* e_edge,
    const unsigned* __restrict__ m_ord, float* __restrict__ denom, int nE)
{
    const int i = blockIdx.x * 256 + threadIdx.x;
    if (i >= nE) return;
    const int d = (int)ei[(size_t)nE + i];
    const float w = __expf(e_edge[i] - ord2f(m_ord[d]));
    e_edge[i] = w;
    atomicAdd(&denom[d], w);
}

// edge pass 3: wave per edge; hagg[d,:] += alpha * h[s,:] (4 floats/lane)
__global__ __launch_bounds__(256) void edge_aggregate(
    const long long* __restrict__ ei, const float* __restrict__ h,
    const float* __restrict__ w_edge, const float* __restrict__ denom,
    float* __restrict__ hagg, int nE)
{
    const int lane = threadIdx.x & 31;
    const int eidx = blockIdx.x * 8 + (threadIdx.x >> 5);
    if (eidx >= nE) return;
    const int s = (int)ei[eidx];
    const int d = (int)ei[(size_t)nE + eidx];
    const float alpha = w_edge[eidx] / fmaxf(denom[d], 1e-16f);
    const float4 hv = *(const float4*)(h + (size_t)s * HD + lane * 4);
    float* o = hagg + (size_t)d * HD + lane * 4;
    atomicAdd(o + 0, alpha * hv.x);
    atomicAdd(o + 1, alpha * hv.y);
    atomicAdd(o + 2, alpha * hv.z);
    atomicAdd(o + 3, alpha * hv.w);
}

// column stats (sum, sumsq) for BN. SMODE 1: v = relu(x + bvec[col]) first.
template<int SMODE>
__global__ __launch_bounds__(128) void col_stats_kernel(
    const float* __restrict__ X, const float* __restrict__ bvec,
    float* __restrict__ sum, float* __restrict__ sumsq, int nrows)
{
    const int col = threadIdx.x;
    float b = 0.f;
    if (SMODE == 1) b = bvec[col];
    float s = 0.f, s2 = 0.f;
    const int r0 = blockIdx.x * 256;
    int r1 = r0 + 256; if (r1 > nrows) r1 = nrows;
    for (int r = r0; r < r1; ++r) {
        float v = X[(size_t)r * HD + col];
        if (SMODE == 1) v = fmaxf(v + b, 0.f);
        s += v; s2 += v * v;
    }
    atomicAdd(&sum[col],   s);
    atomicAdd(&sumsq[col], s2);
}

__global__ void bn_finalize_kernel(
    const float* __restrict__ sum, const float* __restrict__ sumsq,
    const float* __restrict__ gamma, const float* __restrict__ beta,
    float* __restrict__ scale, float* __restrict__ shift, int nrows)
{
    const int c = threadIdx.x;
    const float inv_n = 1.f / (float)nrows;
    const float mu  = sum[c] * inv_n;
    const float var = sumsq[c] * inv_n - mu * mu;
    const float sc  = gamma[c] * rsqrtf(var + 1e-5f);
    scale[c] = sc;
    shift[c] = beta[c] - mu * sc;
}

// row softmax over 32 actions: wave per row, shfl reductions (wave32 native)
__global__ __launch_bounds__(256) void softmax_kernel(
    const float* __restrict__ logits, float* __restrict__ out, int n)
{
    const int lane = threadIdx.x & 31;
    const int row  = blockIdx.x * 8 + (threadIdx.x >> 5);
    if (row >= n) return;
    const float v = logits[(size_t)row * AD + lane];
    float m = v;
#pragma unroll
    for (int off = 16; off > 0; off >>= 1) m = fmaxf(m, __shfl_xor(m, off, 32));
    const float e = __expf(v - m);
    float s = e;
#pragma unroll
    for (int off = 16; off > 0; off >>= 1) s += __shfl_xor(s, off, 32);
    out[(size_t)row * AD + lane] = e / s;
}

// ---------------------------------------------------------------------------
extern "C" void kernel_launch(void* const* d_in, const int* in_sizes, int n_in,
                              void* d_out, int out_size, void* d_ws, size_t ws_size,
                              hipStream_t stream)
{
    (void)in_sizes; (void)n_in; (void)out_size; (void)ws_size;

    const float*     x     = (const float*)d_in[0];
    const long long* ei    = (const long long*)d_in[1];  // int64 per reference
    const float*     W     = (const float*)d_in[2];
    const float*     a_src = (const float*)d_in[3];
    const float*     a_dst = (const float*)d_in[4];
    const float*     b_gat = (const float*)d_in[5];
    const float*     g0    = (const float*)d_in[6];
    const float*     beta0 = (const float*)d_in[7];
    const float*     W1    = (const float*)d_in[8];
    const float*     b1    = (const float*)d_in[9];
    const float*     W2    = (const float*)d_in[10];
    const float*     b2    = (const float*)d_in[11];
    const float*     g2    = (const float*)d_in[12];
    const float*     beta2 = (const float*)d_in[13];
    const float*     W3    = (const float*)d_in[14];
    const float*     b3    = (const float*)d_in[15];
    float* out = (float*)d_out;

    // ---- workspace carve-up (floats) ----
    float* ws = (float*)d_ws;
    float*    h      = ws;                      // NN*HD
    float*    hagg   = h      + (size_t)NN * HD;
    float*    h1     = hagg   + (size_t)NN * HD;
    float*    h2     = h1     + (size_t)NN * HD;
    float*    e_src  = h2     + (size_t)NN * HD;   // NN
    float*    e_dst  = e_src  + NN;
    float*    denom  = e_dst  + NN;
    unsigned* m_ord  = (unsigned*)(denom + NN);    // NN
    float*    e_edge = (float*)(m_ord + NN);       // EE
    float*    logits = e_edge + EE;                // NN*AD
    float*    sum0   = logits + (size_t)NN * AD;   // 8 x 128 stat/scale vectors
    float*    sumsq0 = sum0   + HD;
    float*    scale0 = sumsq0 + HD;
    float*    shift0 = scale0 + HD;
    float*    sum2   = shift0 + HD;
    float*    sumsq2 = sum2   + HD;
    float*    scale2 = sumsq2 + HD;
    float*    shift2 = scale2 + HD;

    // ---- zero accumulation buffers (capture-safe async memsets) ----
    hipMemsetAsync(hagg,  0, (size_t)NN * HD * sizeof(float), stream);
    hipMemsetAsync(denom, 0, (size_t)NN * sizeof(float), stream);
    hipMemsetAsync(m_ord, 0, (size_t)NN * sizeof(unsigned), stream);  // ordered key 0 == minimum
    hipMemsetAsync(sum0,  0, 8 * HD * sizeof(float), stream);

    const int gemmBlocks = (NN + 127) / 128;   // 8 waves * 16 rows per block
    const int nodeBlocks = (NN + 255) / 256;
    const int edgeBlocks = (EE + 255) / 256;
    const int aggBlocks  = (EE + 7) / 8;
    const int statBlocks = (NN + 255) / 256;
    const int smBlocks   = (NN + 7) / 8;

    // 1) h = x @ W                        (N x 256) @ (256 x 128)
    gemm_wmma_kernel<DIN, HD, 0, false><<<gemmBlocks, 256, 0, stream>>>(
        x, W, nullptr, nullptr, nullptr, nullptr, h, NN);

    // 2) per-node attention logits
    node_attn_kernel<<<nodeBlocks, 256, 0, stream>>>(h, a_src, a_dst, e_src, e_dst, NN);

    // 3) edge softmax (max, exp-sum, weighted scatter)
    edge_pass1<<<edgeBlocks, 256, 0, stream>>>(ei, e_src, e_dst, e_edge, m_ord, EE);
    edge_pass2<<<edgeBlocks, 256, 0, stream>>>(ei, e_edge, m_ord, denom, EE);
    edge_aggregate<<<aggBlocks, 256, 0, stream>>>(ei, h, e_edge, denom, hagg, EE);

    // 4) BN0 stats over h0 = relu(hagg + b_gat)
    col_stats_kernel<1><<<statBlocks, 128, 0, stream>>>(hagg, b_gat, sum0, sumsq0, NN);
    bn_finalize_kernel<<<1, 128, 0, stream>>>(sum0, sumsq0, g0, beta0, scale0, shift0, NN);

    // 5) h1 = relu( BN0(relu(hagg+b_gat)) @ W1 + b1 )   (A-transform fused)
    gemm_wmma_kernel<HD, HD, 1, true><<<gemmBlocks, 256, 0, stream>>>(
        hagg, W1, b1, b_gat, scale0, shift0, h1, NN);

    // 6) h2 = relu( h1 @ W2 + b2 )
    gemm_wmma_kernel<HD, HD, 0, true><<<gemmBlocks, 256, 0, stream>>>(
        h1, W2, b2, nullptr, nullptr, nullptr, h2, NN);

    // 7) BN2 stats over h2
    col_stats_kernel<0><<<statBlocks, 128, 0, stream>>>(h2, nullptr, sum2, sumsq2, NN);
    bn_finalize_kernel<<<1, 128, 0, stream>>>(sum2, sumsq2, g2, beta2, scale2, shift2, NN);

    // 8) logits = BN2(h2) @ W3 + b3      (A-transform fused, NCOL = 32)
    gemm_wmma_kernel<HD, AD, 2, false><<<gemmBlocks, 256, 0, stream>>>(
        h2, W3, b3, scale2, shift2, nullptr, logits, NN);

    // 9) row softmax -> output
    softmax_kernel<<<smBlocks, 256, 0, stream>>>(logits, out, NN);
}